// RGRLCore_3590592660274
// MI455X (gfx1250) — compile-verified
//
#include <hip/hip_runtime.h>
#include <math.h>

typedef float v2f __attribute__((ext_vector_type(2)));
typedef float v8f __attribute__((ext_vector_type(8)));
typedef unsigned int u32x4 __attribute__((ext_vector_type(4)));
typedef unsigned int u32x8 __attribute__((ext_vector_type(8)));

// Problem constants (from reference setup_inputs)
#define BS   128   // batch
#define SS   512   // seq len (K)
#define PP   96    // pred len (N)
#define CC   862   // channels
#define EPSF 1e-5f

// ---------------------------------------------------------------------------
// Kernel 1: fused RevIN stats + normalize + transpose (unchanged from R1).
// Reads x[B,S,C] once (64B-coalesced across a 16-channel block through LDS),
// computes per-(b,c) mean / biased std, folds RevIN affine into xn = a*x + be,
// writes xnT[c][b][s] (s-contiguous) for the per-channel GEMM.
// ---------------------------------------------------------------------------
__global__ __launch_bounds__(256) void revin_transpose_kernel(
    const float* __restrict__ x,
    const float* __restrict__ rev_w,
    const float* __restrict__ rev_b,
    float* __restrict__ xnT,
    float* __restrict__ meanWS,
    float* __restrict__ stdWS)
{
    __shared__ float tile[16 * 513];
    __shared__ float psum[256];
    __shared__ float psq[256];
    __shared__ float alph[16];
    __shared__ float bet[16];

    const int t  = threadIdx.x;
    const int cl = t & 15;
    const int si = t >> 4;
    const int c0 = blockIdx.x * 16;
    const int b  = blockIdx.y;
    const int c  = c0 + cl;
    const bool cvalid = (c < CC);

    float s1 = 0.f, s2 = 0.f;
    #pragma unroll 4
    for (int k = 0; k < 32; ++k) {
        const int s = si + (k << 4);
        const float v = cvalid ? x[((size_t)b * SS + s) * CC + c] : 0.f;
        tile[cl * 513 + s] = v;
        s1 += v;
        s2 += v * v;
    }
    psum[cl * 16 + si] = s1;
    psq [cl * 16 + si] = s2;
    __syncthreads();

    if (t < 16) {
        const int cc = c0 + t;
        float sum = 0.f, sq = 0.f;
        #pragma unroll
        for (int i = 0; i < 16; ++i) { sum += psum[t * 16 + i]; sq += psq[t * 16 + i]; }
        const float mean = sum * (1.0f / SS);
        float var = sq * (1.0f / SS) - mean * mean;
        var = var < 0.f ? 0.f : var;
        const float sd = sqrtf(var) + EPSF;
        float a = 0.f, be = 0.f;
        if (cc < CC) {
            a  = rev_w[cc] / sd;
            be = rev_b[cc] - mean * a;
            meanWS[(size_t)b * CC + cc] = mean;
            stdWS [(size_t)b * CC + cc] = sd;
        }
        alph[t] = a;
        bet[t]  = be;
    }
    __syncthreads();

    #pragma unroll 4
    for (int r = 0; r < 16; ++r) {
        if (c0 + r < CC) {
            const float a = alph[r], be = bet[r];
            float* dst = xnT + ((size_t)(c0 + r) * BS + b) * SS;
            dst[t]       = a * tile[r * 513 + t]       + be;
            dst[t + 256] = a * tile[r * 513 + t + 256] + be;
        }
    }
}

// ---------------------------------------------------------------------------
// TDM: issue one 2D tensor_load_to_lds of a rows x 32-float tile whose rows
// are SS apart in global memory, with hardware LDS padding of 4 dwords per
// 32-dword row (-> LDS row stride 36 dwords, bank-conflict-free frag reads).
//   D# group0: count=1, lds_addr, global_addr(57b), type=2
//   D# group1: data_size=4B, pad_enable, pad_interval=4 (128B), pad_amount=3
//              (4 dwords), tensor/tile dims = rows x 32, dim0_stride = 512
// ---------------------------------------------------------------------------
#define KB  32
#define LDA 36   // KB + 4 pad dwords

__device__ __forceinline__ void tdm_load_tile(const float* gptr, unsigned ldsByte,
                                              unsigned rows)
{
    const unsigned long long ga = (unsigned long long)(uintptr_t)gptr;
    u32x4 g0;
    g0[0] = 1u;                                                // count=1
    g0[1] = ldsByte;                                           // lds_addr
    g0[2] = (unsigned)ga;                                      // global_addr[31:0]
    g0[3] = ((unsigned)(ga >> 32) & 0x01FFFFFFu) | (2u << 30); // ga[56:32] | type=2
    u32x8 g1;
    g1[0] = (2u << 16) | (1u << 20) | (4u << 22) | (3u << 25); // 4B, pad 4dw/128B
    g1[1] = (unsigned)KB << 16;                                // tensor_dim0 = 32
    g1[2] = rows << 16;                                        // tensor_dim1 = rows
    g1[3] = (unsigned)KB << 16;                                // tile_dim0 = 32
    g1[4] = rows;                                              // tile_dim1, tile_dim2=0
    g1[5] = (unsigned)SS;                                      // dim0_stride = 512
    g1[6] = 0u;
    g1[7] = 0u;
    asm volatile("tensor_load_to_lds %0, %1" :: "s"(g0), "s"(g1) : "memory");
}

// ---------------------------------------------------------------------------
// Kernel 2: per-channel GEMM Y[128,96] = XnT[c] (128x512) * W[c]^T, f32 WMMA.
// 8 waves as 4(M) x 2(N): each wave 32 rows x 48 cols -> 6 v8f accumulators,
// 2 A-frags + 3 B-frags per K-step (5 LDS loads : 6 WMMAs).
// Double-buffered LDS fed by TDM (wave 0 issues, TENSORcnt-pipelined), so the
// DMA of block n+1 overlaps WMMA on block n. Fused bias + RevIN denorm.
// ---------------------------------------------------------------------------
__global__ __launch_bounds__(256) void channel_gemm_wmma_kernel(
    const float* __restrict__ W,
    const float* __restrict__ bias,
    const float* __restrict__ rev_w,
    const float* __restrict__ rev_b,
    const float* __restrict__ xnT,
    const float* __restrict__ meanWS,
    const float* __restrict__ stdWS,
    float* __restrict__ out)
{
    __shared__ float smem[2][(BS + PP) * LDA];   // 2 x 224 x 36 x 4B = 64,512 B

    const int c    = blockIdx.x;
    const int t    = threadIdx.x;
    const int lane = t & 31;           // wave32
    const int wave = t >> 5;           // 0..7
    const int mw   = wave & 3;         // 4 M strips
    const int nw   = wave >> 2;        // 2 N halves
    const int m0   = mw * 32;
    const int n0   = nw * 48;
    const int l15  = lane & 15;
    const int lhi  = lane >> 4;        // K sub-pair for A/B frags, M half for C/D

    const float* Ac = xnT + (size_t)c * BS * SS;
    const float* Wc = W   + (size_t)c * PP * SS;

    const unsigned ldsA[2] = { (unsigned)(uintptr_t)&smem[0][0],
                               (unsigned)(uintptr_t)&smem[1][0] };
    const unsigned bOff = (unsigned)(BS * LDA * 4);   // B tile after 128 A rows

    v8f acc[2][3];
    #pragma unroll
    for (int ai = 0; ai < 2; ++ai)
        #pragma unroll
        for (int j = 0; j < 3; ++j) acc[ai][j] = (v8f)0.0f;

    constexpr int NBLK = SS / KB;      // 16
    const bool tdmWave = (t < 32);     // only wave 0 drives the TDM

    if (tdmWave) {                     // prefetch block 0 into buffer 0
        tdm_load_tile(Ac, ldsA[0],        BS);
        tdm_load_tile(Wc, ldsA[0] + bOff, PP);
    }

    for (int n = 0; n < NBLK; ++n) {
        if (tdmWave) {
            if (n + 1 < NBLK) {        // stream block n+1 into the other buffer
                const int kb = (n + 1) * KB;
                const unsigned dst = ldsA[(n + 1) & 1];
                tdm_load_tile(Ac + kb, dst,        BS);
                tdm_load_tile(Wc + kb, dst + bOff, PP);
                __builtin_amdgcn_s_wait_tensorcnt((short)2);  // block n resident
            } else {
                __builtin_amdgcn_s_wait_tensorcnt((short)0);
            }
        }
        __syncthreads();               // LDS block n visible to all waves

        const float* As = &smem[n & 1][0];
        const float* Bs = As + BS * LDA;

        #pragma unroll
        for (int kk = 0; kk < KB; kk += 4) {
            const v2f a0 = *(const v2f*)&As[(m0      + l15) * LDA + kk + lhi * 2];
            const v2f a1 = *(const v2f*)&As[(m0 + 16 + l15) * LDA + kk + lhi * 2];
            #pragma unroll
            for (int j = 0; j < 3; ++j) {
                const v2f bf = *(const v2f*)&Bs[(n0 + j * 16 + l15) * LDA + kk + lhi * 2];
                acc[0][j] = __builtin_amdgcn_wmma_f32_16x16x4_f32(
                    false, a0, false, bf, (short)0, acc[0][j], false, false);
                acc[1][j] = __builtin_amdgcn_wmma_f32_16x16x4_f32(
                    false, a1, false, bf, (short)0, acc[1][j], false, false);
            }
        }
        __syncthreads();               // protect buffer before TDM reuses it
    }

    // Epilogue: y = (acc + bias - rev_b) / (rev_w + eps) * std + mean
    const float revb_c = rev_b[c];
    const float inv_rw = 1.0f / (rev_w[c] + EPSF);
    #pragma unroll
    for (int ai = 0; ai < 2; ++ai) {
        #pragma unroll
        for (int j = 0; j < 3; ++j) {
            const int p  = n0 + j * 16 + l15;
            const float bp = bias[(size_t)c * PP + p];
            #pragma unroll
            for (int r = 0; r < 8; ++r) {
                const int bg = m0 + ai * 16 + lhi * 8 + r;
                const float sd = stdWS [(size_t)bg * CC + c];
                const float mn = meanWS[(size_t)bg * CC + c];
                const float y  = (acc[ai][j][r] + bp - revb_c) * inv_rw * sd + mn;
                out[(size_t)bg * (PP * CC) + (size_t)p * CC + c] = y;
            }
        }
    }
}

// ---------------------------------------------------------------------------
// Host-side launcher
// Inputs: x[B,S,C], W[C,P,S], b[C,P], rev_w[C], rev_b[C] (all f32).
// Output: y[B,P,C] f32.  Workspace: xnT | mean | std (~227 MB).
// ---------------------------------------------------------------------------
extern "C" void kernel_launch(void* const* d_in, const int* in_sizes, int n_in,
                              void* d_out, int out_size, void* d_ws, size_t ws_size,
                              hipStream_t stream) {
    const float* x    = (const float*)d_in[0];
    const float* W    = (const float*)d_in[1];
    const float* bias = (const float*)d_in[2];
    const float* rvw  = (const float*)d_in[3];
    const float* rvb  = (const float*)d_in[4];
    float* out = (float*)d_out;

    float* ws     = (float*)d_ws;
    float* xnT    = ws;
    float* meanWS = ws + (size_t)CC * BS * SS;
    float* stdWS  = meanWS + (size_t)BS * CC;

    dim3 g1((CC + 15) / 16, BS);
    revin_transpose_kernel<<<g1, 256, 0, stream>>>(x, rvw, rvb, xnT, meanWS, stdWS);

    dim3 g2(CC);
    channel_gemm_wmma_kernel<<<g2, 256, 0, stream>>>(W, bias, rvw, rvb,
                                                     xnT, meanWS, stdWS, out);
}